// BidirectionalMambaLayer_78314433675554
// MI455X (gfx1250) — compile-verified
//
#include <hip/hip_runtime.h>
#include <hip/hip_bf16.h>
#include <math.h>
#include <stdint.h>

// ---------------------------------------------------------------------------
// Bidirectional Mamba layer for gfx1250 (MI455X), fp32 end-to-end.
// GEMMs run on the WMMA pipe via V_WMMA_F32_16X16X4_F32 with a double-buffered
// LDS pipeline: K-chunks staged by GLOBAL_LOAD_ASYNC_TO_LDS_B128 (ASYNCcnt)
// while WMMAs consume the previous chunk; fragments fed by ds_load_b64.
// ---------------------------------------------------------------------------

#define BATCH    2
#define SEQ      1024
#define DMODEL   1024
#define DINNER   2048
#define DCONV    4
#define DTRANK   64
#define DSTATE   16
#define ROWS     (BATCH * SEQ)          // 2048
#define XDBL_W   (DTRANK + 2 * DSTATE)  // 96

#define KC       16                     // K-chunk staged per pipeline stage
#define LPAD     20                     // LDS row pitch (floats): 16B-aligned
                                        // b128 stores + conflict-free b64 reads

#define AS_GLOBAL __attribute__((address_space(1)))
#define AS_LOCAL  __attribute__((address_space(3)))

#if defined(__has_builtin)
#  if __has_builtin(__builtin_amdgcn_global_load_async_to_lds_b128)
#    define USE_ASYNC_LDS 1
#  endif
#endif
#ifndef USE_ASYNC_LDS
#  define USE_ASYNC_LDS 0
#endif

typedef __attribute__((ext_vector_type(2))) float v2f;
typedef __attribute__((ext_vector_type(8))) float v8f;
typedef int v4i_gld __attribute__((vector_size(16)));   // builtin's pointee type

__device__ __forceinline__ v8f wmma_f32_k4(v2f a, v2f b, v8f c) {
    // D = A(16x4) x B(4x16) + C(16x16), fp32
    return __builtin_amdgcn_wmma_f32_16x16x4_f32(
        /*neg_a=*/false, a, /*neg_b=*/false, b,
        /*c_mod=*/(short)0, c, /*reuse_a=*/false, /*reuse_b=*/false);
}

__device__ __forceinline__ float softplus_f(float x) {
    return (x > 20.0f) ? x : log1pf(__expf(x));
}

__device__ __forceinline__ float silu_f(float x) {
    return x * (1.0f / (1.0f + __expf(-x)));
}

// 16-byte global -> LDS copy: async DMA path if available, VGPR fallback else.
#if USE_ASYNC_LDS
#define GLD16(gp, lp)                                                         \
    __builtin_amdgcn_global_load_async_to_lds_b128(                           \
        (AS_GLOBAL v4i_gld*)(uintptr_t)(gp), (AS_LOCAL v4i_gld*)(lp), 0, 0)
__device__ __forceinline__ void wait_stage(bool more) {
#  if __has_builtin(__builtin_amdgcn_s_wait_asynccnt)
    if (more) __builtin_amdgcn_s_wait_asynccnt(4);
    else      __builtin_amdgcn_s_wait_asynccnt(0);
#  else
    if (more) asm volatile("s_wait_asynccnt 4" ::: "memory");
    else      asm volatile("s_wait_asynccnt 0" ::: "memory");
#  endif
}
#else
#define GLD16(gp, lp) (*(float4*)(lp) = *(const float4*)(gp))
__device__ __forceinline__ void wait_stage(bool) {}
#endif

// ---------------------------------------------------------------------------
// LDS-tiled fp32 WMMA GEMM:  C[M,N] = epi( A[M,K] * W[N,K]^T + bias )
//   block tile : 128(M) x 128(N), 8 waves as 2(M) x 4(N)
//   wave tile  : 64(M) x 32(N)  -> 8 v8f accumulators
//   K pipelined in chunks of 16 through double-buffered LDS (4 x 10 KB)
// flipA / flipC reverse the time index (low 10 bits) of the row address,
// implementing the backward direction's sequence flip for free.
// EPI: 0 = none, 1 = +bias, 2 = softplus(x + bias)
// ---------------------------------------------------------------------------
template <int EPI>
__global__ __launch_bounds__(256)
void gemm_wmma_lds(const float* __restrict__ A, int lda, int flipA,
                   const float* __restrict__ W, int ldw,
                   const float* __restrict__ bias,
                   float* __restrict__ C, int ldc, int flipC,
                   int M, int N, int K, int nBlocksN) {
    __shared__ float lA[2][128 * LPAD];
    __shared__ float lB[2][128 * LPAD];

    const int bm = blockIdx.x / nBlocksN;
    const int bn = blockIdx.x % nBlocksN;
    const int mBlock = bm << 7;
    const int nBlock = bn << 7;

    const int tid  = threadIdx.x;
    const int wid  = tid >> 5;
    const int lane = tid & 31;
    const int wm   = wid >> 2;          // 0..1 : wave's 64-row strip
    const int wn   = wid & 3;           // 0..3 : wave's 32-col strip
    const int half = lane >> 4;         // K-pair selector (WMMA A/B layout)
    const int lm   = lane & 15;         // row/col within 16-wide tile

    // staging mapping: 4 threads cover one 16-float (64B) row chunk
    const int sr = tid >> 2;            // row 0..63 (+64 for second half)
    const int sq = tid & 3;             // float4 index within the row chunk

    // per-thread global staging pointers (2 rows each for A and W)
    const float* pAg[2];
    const float* pWg[2];
#pragma unroll
    for (int it = 0; it < 2; ++it) {
        const int r  = sr + it * 64;
        const int mg = mBlock + r;
        const int rowA = flipA ? (mg ^ (SEQ - 1)) : mg;
        pAg[it] = A + (size_t)rowA * lda + sq * 4;
        int ng = nBlock + r;
        if (ng >= N) ng = N - 1;        // clamp (only hit when N % 128 != 0)
        pWg[it] = W + (size_t)ng * ldw + sq * 4;
    }

#define STAGE(pbuf, kofs)                                                     \
    do {                                                                      \
        GLD16(pAg[0] + (kofs), &lA[(pbuf)][(sr)      * LPAD + sq * 4]);       \
        GLD16(pWg[0] + (kofs), &lB[(pbuf)][(sr)      * LPAD + sq * 4]);       \
        GLD16(pAg[1] + (kofs), &lA[(pbuf)][(sr + 64) * LPAD + sq * 4]);       \
        GLD16(pWg[1] + (kofs), &lB[(pbuf)][(sr + 64) * LPAD + sq * 4]);       \
    } while (0)

    v8f acc[4][2];
#pragma unroll
    for (int i = 0; i < 4; ++i) {
        acc[i][0] = (v8f){};
        acc[i][1] = (v8f){};
    }

    // ---- software pipeline: stage chunk i+1 while computing chunk i ----
    STAGE(0, 0);
    int par = 0;
    for (int k0 = 0; k0 < K; k0 += KC) {
        const bool more = (k0 + KC) < K;
        if (more) STAGE(par ^ 1, k0 + KC);
        wait_stage(more);               // own older async copies have landed
        __syncthreads();                // everyone's copies have landed

        const float* pa0 = &lA[par][(wm * 64 + lm) * LPAD + 2 * half];
        const float* pb0 = &lB[par][(wn * 32 + lm) * LPAD + 2 * half];
#pragma unroll
        for (int kk = 0; kk < KC; kk += 4) {
            v2f b0 = *(const v2f*)(pb0 + kk);
            v2f b1 = *(const v2f*)(pb0 + 16 * LPAD + kk);
#pragma unroll
            for (int i = 0; i < 4; ++i) {
                v2f a = *(const v2f*)(pa0 + i * 16 * LPAD + kk);
                acc[i][0] = wmma_f32_k4(a, b0, acc[i][0]);
                acc[i][1] = wmma_f32_k4(a, b1, acc[i][1]);
            }
        }
        __syncthreads();                // buffer free for the next stage
        par ^= 1;
    }
#undef STAGE

    // ---- epilogue + store ----
    // C/D layout: reg r -> lanes0-15 hold M=r, lanes16-31 hold M=r+8
#pragma unroll
    for (int i = 0; i < 4; ++i) {
#pragma unroll
        for (int j = 0; j < 2; ++j) {
            const int nOut = nBlock + wn * 32 + 16 * j + lm;
            if (nOut < N) {
                const float bv = (EPI > 0) ? bias[nOut] : 0.0f;
#pragma unroll
                for (int r = 0; r < 8; ++r) {
                    int m = mBlock + wm * 64 + 16 * i + half * 8 + r;
                    float v = acc[i][j][r] + bv;
                    if (EPI == 2) v = softplus_f(v);
                    int row = flipC ? (m ^ (SEQ - 1)) : m;
                    C[(size_t)row * ldc + nOut] = v;
                }
            }
        }
    }
}

// ---------------------------------------------------------------------------
// Depthwise causal conv (width 4) + bias + SiLU.
// xz rows are (b*SEQ+t) with width 2*DINNER; xs = first DINNER columns.
// ---------------------------------------------------------------------------
__global__ __launch_bounds__(256)
void conv_silu_kernel(const float* __restrict__ xz,
                      const float* __restrict__ convw,
                      const float* __restrict__ convb,
                      float* __restrict__ u) {
    const int idx = blockIdx.x * blockDim.x + threadIdx.x;
    if (idx >= ROWS * DINNER) return;
    const int c = idx & (DINNER - 1);
    const int r = idx >> 11;                 // row = b*SEQ + t
    const int t = r & (SEQ - 1);

    const float w0 = convw[c * 4 + 0];
    const float w1 = convw[c * 4 + 1];
    const float w2 = convw[c * 4 + 2];
    const float w3 = convw[c * 4 + 3];

    const float* col = xz + (size_t)r * (2 * DINNER) + c;
    float acc = convb[c];
    if (t >= 3) acc += w0 * col[-(ptrdiff_t)3 * 2 * DINNER];
    if (t >= 2) acc += w1 * col[-(ptrdiff_t)2 * 2 * DINNER];
    if (t >= 1) acc += w2 * col[-(ptrdiff_t)1 * 2 * DINNER];
    acc += w3 * col[0];

    u[(size_t)r * DINNER + c] = silu_f(acc);
}

// ---------------------------------------------------------------------------
// Selective scan. One lane per (batch, channel): 16 states in registers,
// 1024 sequential steps. B/C (16 floats each) loaded as 4x float4 broadcasts.
// Fused epilogue: y = (scan + u*D) * silu(z).
// ---------------------------------------------------------------------------
__global__ __launch_bounds__(256)
void scan_kernel(const float* __restrict__ dt,
                 const float* __restrict__ u,
                 const float* __restrict__ xz,     // z = columns [DINNER, 2*DINNER)
                 const float* __restrict__ xdbl,   // [ROWS, 96]
                 const float* __restrict__ Alog,   // [DINNER, 16]
                 const float* __restrict__ Dp,     // [DINNER]
                 float* __restrict__ ymod) {
    const int idx = blockIdx.x * blockDim.x + threadIdx.x;
    if (idx >= BATCH * DINNER) return;
    const int c = idx & (DINNER - 1);
    const int b = idx >> 11;

    float Aval[DSTATE], h[DSTATE];
#pragma unroll
    for (int s = 0; s < DSTATE; ++s) {
        Aval[s] = -__expf(Alog[c * DSTATE + s]);
        h[s] = 0.0f;
    }
    const float dval = Dp[c];

    for (int t = 0; t < SEQ; ++t) {
        const size_t r = (size_t)b * SEQ + t;
        const float dtv = dt[r * DINNER + c];
        const float uv  = u[r * DINNER + c];
        const float zv  = xz[r * (2 * DINNER) + DINNER + c];

        float Bs[DSTATE], Cs[DSTATE];
        const float4* pB = (const float4*)(xdbl + r * XDBL_W + DTRANK);
        const float4* pC = (const float4*)(xdbl + r * XDBL_W + DTRANK + DSTATE);
#pragma unroll
        for (int i = 0; i < 4; ++i) {
            float4 qb = pB[i];
            float4 qc = pC[i];
            Bs[4 * i + 0] = qb.x; Bs[4 * i + 1] = qb.y;
            Bs[4 * i + 2] = qb.z; Bs[4 * i + 3] = qb.w;
            Cs[4 * i + 0] = qc.x; Cs[4 * i + 1] = qc.y;
            Cs[4 * i + 2] = qc.z; Cs[4 * i + 3] = qc.w;
        }

        const float du = dtv * uv;
        float acc = 0.0f;
#pragma unroll
        for (int s = 0; s < DSTATE; ++s) {
            h[s] = fmaf(h[s], __expf(dtv * Aval[s]), du * Bs[s]);
            acc  = fmaf(h[s], Cs[s], acc);
        }
        const float y = acc + uv * dval;
        ymod[r * DINNER + c] = y * silu_f(zv);
    }
}

// ---------------------------------------------------------------------------
// Host-side launch
// ---------------------------------------------------------------------------
static inline void launch_gemm(int epi,
                               const float* A, int lda, int flipA,
                               const float* W, int ldw, const float* bias,
                               float* C, int ldc, int flipC,
                               int M, int N, int K, hipStream_t stream) {
    const int nBlocksN = (N + 127) / 128;
    const int blocks   = (M / 128) * nBlocksN;
    if (epi == 0)
        gemm_wmma_lds<0><<<blocks, 256, 0, stream>>>(A, lda, flipA, W, ldw, bias, C, ldc, flipC, M, N, K, nBlocksN);
    else if (epi == 1)
        gemm_wmma_lds<1><<<blocks, 256, 0, stream>>>(A, lda, flipA, W, ldw, bias, C, ldc, flipC, M, N, K, nBlocksN);
    else
        gemm_wmma_lds<2><<<blocks, 256, 0, stream>>>(A, lda, flipA, W, ldw, bias, C, ldc, flipC, M, N, K, nBlocksN);
}

extern "C" void kernel_launch(void* const* d_in, const int* in_sizes, int n_in,
                              void* d_out, int out_size, void* d_ws, size_t ws_size,
                              hipStream_t stream) {
    (void)in_sizes; (void)n_in; (void)out_size; (void)ws_size;

    const float* x = (const float*)d_in[0];
    const float* Win[2]   = {(const float*)d_in[1],  (const float*)d_in[10]};
    const float* convw[2] = {(const float*)d_in[2],  (const float*)d_in[11]};
    const float* convb[2] = {(const float*)d_in[3],  (const float*)d_in[12]};
    const float* Wx[2]    = {(const float*)d_in[4],  (const float*)d_in[13]};
    const float* Wdt[2]   = {(const float*)d_in[5],  (const float*)d_in[14]};
    const float* bdt[2]   = {(const float*)d_in[6],  (const float*)d_in[15]};
    const float* Alog[2]  = {(const float*)d_in[7],  (const float*)d_in[16]};
    const float* Dpar[2]  = {(const float*)d_in[8],  (const float*)d_in[17]};
    const float* Wout[2]  = {(const float*)d_in[9],  (const float*)d_in[18]};
    const float* fuseW = (const float*)d_in[19];
    const float* fuseb = (const float*)d_in[20];
    float* out = (float*)d_out;

    // Workspace carve-up (floats); total ~101 MB.
    float* ws   = (float*)d_ws;
    float* cat  = ws;                                   // ROWS * 2*DMODEL
    float* xz   = cat  + (size_t)ROWS * 2 * DMODEL;     // ROWS * 2*DINNER
    float* u    = xz   + (size_t)ROWS * 2 * DINNER;     // ROWS * DINNER
    float* xdbl = u    + (size_t)ROWS * DINNER;         // ROWS * 96
    float* dt   = xdbl + (size_t)ROWS * XDBL_W;         // ROWS * DINNER
    float* ym   = dt   + (size_t)ROWS * DINNER;         // ROWS * DINNER

    const int convThreads = ROWS * DINNER;

    for (int d = 0; d < 2; ++d) {
        const int flip = d;   // direction 1 = backward (time-reversed)

        // xz = x(flip) @ Win^T                      [2048 x 4096, K=1024]
        launch_gemm(0, x, DMODEL, flip, Win[d], DMODEL, nullptr,
                    xz, 2 * DINNER, 0, ROWS, 2 * DINNER, DMODEL, stream);

        // u = silu(depthwise_conv(xs) + convb)
        conv_silu_kernel<<<convThreads / 256, 256, 0, stream>>>(xz, convw[d], convb[d], u);

        // x_dbl = u @ Wx^T                          [2048 x 96, K=2048]
        launch_gemm(0, u, DINNER, 0, Wx[d], DINNER, nullptr,
                    xdbl, XDBL_W, 0, ROWS, XDBL_W, DINNER, stream);

        // dt = softplus(dtp @ Wdt^T + bdt)          [2048 x 2048, K=64]
        launch_gemm(2, xdbl, XDBL_W, 0, Wdt[d], DTRANK, bdt[d],
                    dt, DINNER, 0, ROWS, DINNER, DTRANK, stream);

        // selective scan + gating epilogue -> ym
        scan_kernel<<<(BATCH * DINNER) / 256, 256, 0, stream>>>(
            dt, u, xz, xdbl, Alog[d], Dpar[d], ym);

        // cat[:, d*1024:(d+1)*1024] = ym(flip back) @ Wout^T   [2048 x 1024, K=2048]
        launch_gemm(0, ym, DINNER, 0, Wout[d], DINNER, nullptr,
                    cat + (size_t)d * DMODEL, 2 * DMODEL, flip,
                    ROWS, DMODEL, DINNER, stream);
    }

    // out = cat @ fuseW^T + fuseb                   [2048 x 1024, K=2048]
    launch_gemm(1, cat, 2 * DMODEL, 0, fuseW, 2 * DMODEL, fuseb,
                out, DMODEL, 0, ROWS, DMODEL, 2 * DMODEL, stream);
}